// PSRoIAlignRotated_2327872274503
// MI455X (gfx1250) — compile-verified
//
#include <hip/hip_runtime.h>
#include <hip/hip_bf16.h>
#include <stdint.h>

// Problem constants (from the reference)
#define OUT_P   7          // OUT_SIZE
#define GRP     7          // GROUP_SIZE
#define NDIM    10         // OUTPUT_DIM
#define NSAMP   2          // SAMPLE_NUM
#define FH      160
#define FW      160
#define NCH     (NDIM * GRP * GRP)   // 490; chan(d,py,px) == d*49+py*7+px == flat e
#define SCALEF  0.125f

__global__ __launch_bounds__(256) void psroi_rot_kernel(
    const float* __restrict__ feat,
    const float* __restrict__ rois,
    float* __restrict__ out,
    int R)
{
    __shared__ float s_roi[8];

    const int r = blockIdx.x;
    if (r >= R) return;

    // --- CDNA5 async global->LDS broadcast of the 6 ROI floats -------------
    // 6 lanes copy one dword each straight into LDS (ASYNCcnt path); the
    // issuing wave drains ASYNCcnt, the split barrier publishes to the block.
    if (threadIdx.x < 6) {
        const float* gp = rois + (size_t)r * 6 + threadIdx.x;
        unsigned lds_off = (unsigned)(uintptr_t)&s_roi[threadIdx.x];
        asm volatile("global_load_async_to_lds_b32 %0, %1, off"
                     :: "v"(lds_off), "v"(gp)
                     : "memory");
    }
    asm volatile("s_wait_asynccnt 0x0" ::: "memory");
    __syncthreads();

    const int   b     = (int)s_roi[0];
    const float cx    = s_roi[1] * SCALEF;
    const float cy    = s_roi[2] * SCALEF;
    const float rw    = fmaxf(s_roi[3] * SCALEF, 0.1f);
    const float rh    = fmaxf(s_roi[4] * SCALEF, 0.1f);
    const float theta = s_roi[5];

    const float sinT = sinf(theta);
    const float cosT = cosf(theta);

    const float bin_h = rh * (1.0f / OUT_P);
    const float bin_w = rw * (1.0f / OUT_P);

    const size_t bbase = (size_t)b * NCH * (size_t)(FH * FW);

    // 490 output elements per ROI; element index == channel index.
    for (int e = threadIdx.x; e < NCH; e += blockDim.x) {
        const int py = (e / OUT_P) % OUT_P;
        const int px = e % OUT_P;
        const float* __restrict__ fp = feat + bbase + (size_t)e * (FH * FW);

        float acc = 0.0f;
        #pragma unroll
        for (int sy = 0; sy < NSAMP; ++sy) {
            const float yy = -0.5f * rh +
                ((float)py + ((float)sy + 0.5f) * (1.0f / NSAMP)) * bin_h;
            #pragma unroll
            for (int sx = 0; sx < NSAMP; ++sx) {
                const float xx = -0.5f * rw +
                    ((float)px + ((float)sx + 0.5f) * (1.0f / NSAMP)) * bin_w;

                const float x = cosT * xx - sinT * yy + cx;
                const float y = sinT * xx + cosT * yy + cy;

                // Branchless validity: clamped indices are always in-range,
                // so issue all gathers unconditionally (max MLP on the
                // L2-latency-bound path) and mask the contribution.
                const bool valid =
                    (y > -1.0f) && (y < (float)FH) &&
                    (x > -1.0f) && (x < (float)FW);
                const float vmask = valid ? 1.0f : 0.0f;

                const float yc  = fminf(fmaxf(y, 0.0f), (float)(FH - 1));
                const float xc  = fminf(fmaxf(x, 0.0f), (float)(FW - 1));
                const float y0f = floorf(yc);
                const float x0f = floorf(xc);
                const int   y0  = (int)y0f;
                const int   x0  = (int)x0f;
                const int   y1  = min(y0 + 1, FH - 1);
                const int   x1  = min(x0 + 1, FW - 1);
                const float ly  = yc - y0f;
                const float lx  = xc - x0f;
                const float hy  = 1.0f - ly;
                const float hx  = 1.0f - lx;

                // 4 independent gathers, no branch around them: all 16 loads
                // of this element pipeline on LOADcnt; L2-resident features.
                const float v00 = fp[y0 * FW + x0];
                const float v01 = fp[y0 * FW + x1];
                const float v10 = fp[y1 * FW + x0];
                const float v11 = fp[y1 * FW + x1];

                acc += vmask * (hy * hx * v00 + hy * lx * v01 +
                                ly * hx * v10 + ly * lx * v11);
            }
        }
        // Write-once output: non-temporal store keeps L2 for the features.
        __builtin_nontemporal_store(acc * (1.0f / (NSAMP * NSAMP)),
                                    &out[(size_t)r * NCH + e]);
    }
}

extern "C" void kernel_launch(void* const* d_in, const int* in_sizes, int n_in,
                              void* d_out, int out_size, void* d_ws, size_t ws_size,
                              hipStream_t stream) {
    (void)n_in; (void)out_size; (void)d_ws; (void)ws_size;
    const float* feat = (const float*)d_in[0];
    const float* rois = (const float*)d_in[1];
    float*       out  = (float*)d_out;

    const int R = in_sizes[1] / 6;   // 512 ROIs

    psroi_rot_kernel<<<dim3(R), dim3(256), 0, stream>>>(feat, rois, out, R);
}